// CondNetMetrics_30021821399478
// MI455X (gfx1250) — compile-verified
//
#include <hip/hip_runtime.h>
#include <hip/hip_bf16.h>

#define NN 65536
#define PP 128

typedef __attribute__((ext_vector_type(2))) float v2f;
typedef __attribute__((ext_vector_type(8))) float v8f;

// Output layout (floats):
// rms[128]@0, rmsq[128]@128, Np[128]@256, nn[128*5]@384,
// rms_g@1024, rmsq_g@1025, Ng@1026, nn_g[5]@1027, db@1032,
// N_node[65536]@1033, dist_edge[128*65536]@66569, belongs_edge@8455177

// Workspace layout (floats):
// [0..127] Np, [128..255] S2 = sum ||x||^2 over members,
// [256..639] S1 = sum x over members (3 per p),
// [640] sum q, [641] sum q^2, [642] sum q^2*||x||^2,
// [643..645] sum q^2*x, [646] sum ||x||^2

// ---------- Kernel 1: scatter sums over nodes ----------
__global__ __launch_bounds__(256) void cnm_reduce(
    const float* __restrict__ x, const float* __restrict__ q,
    const int* __restrict__ parent, float* __restrict__ ws)
{
    __shared__ float acc[656];
    for (int i = threadIdx.x; i < 656; i += 256) acc[i] = 0.0f;
    __syncthreads();

    int n = blockIdx.x * 256 + threadIdx.x;           // exactly covers 65536
    float x0 = x[3*n], x1 = x[3*n+1], x2 = x[3*n+2];
    float xn2 = x0*x0 + x1*x1 + x2*x2;
    float qv = q[n], q2 = qv*qv;
    int p = parent[n];

    atomicAdd(&acc[p], 1.0f);
    atomicAdd(&acc[128 + p], xn2);
    atomicAdd(&acc[256 + 3*p + 0], x0);
    atomicAdd(&acc[256 + 3*p + 1], x1);
    atomicAdd(&acc[256 + 3*p + 2], x2);
    atomicAdd(&acc[640], qv);
    atomicAdd(&acc[641], q2);
    atomicAdd(&acc[642], q2 * xn2);
    atomicAdd(&acc[643], q2 * x0);
    atomicAdd(&acc[644], q2 * x1);
    atomicAdd(&acc[645], q2 * x2);
    atomicAdd(&acc[646], xn2);
    __syncthreads();

    for (int i = threadIdx.x; i < 647; i += 256)
        atomicAdd(&ws[i], acc[i]);
}

// ---------- Kernel 2: top-5 smallest cond distances (129 receivers) ----------
__global__ __launch_bounds__(256) void cnm_topk(
    const float* __restrict__ x, const float* __restrict__ is_cond,
    const float* __restrict__ mx, float* __restrict__ out)
{
    int b = blockIdx.x;                                // 0..127 particles, 128 = global
    float c0 = 0.f, c1 = 0.f, c2 = 0.f;
    if (b < PP) { c0 = mx[3*b]; c1 = mx[3*b+1]; c2 = mx[3*b+2]; }

    float best[5] = {3.0e38f, 3.0e38f, 3.0e38f, 3.0e38f, 3.0e38f};
    for (int n = threadIdx.x; n < NN; n += 256) {
        float d0 = x[3*n] - c0, d1 = x[3*n+1] - c1, d2 = x[3*n+2] - c2;
        float dist = __builtin_amdgcn_sqrtf(d0*d0 + d1*d1 + d2*d2);
        float c = dist * is_cond[n];
        float v = (c < 1e-8f) ? 999.0f : c;
        if (v < best[4]) {
            best[4] = v;
            #pragma unroll
            for (int k = 4; k > 0; --k)
                if (best[k] < best[k-1]) { float t = best[k]; best[k] = best[k-1]; best[k-1] = t; }
        }
    }

    __shared__ float buf[1280];
    __shared__ float rmin[256];
    __shared__ int   ridx[256];
    #pragma unroll
    for (int k = 0; k < 5; ++k) buf[threadIdx.x * 5 + k] = best[k];
    __syncthreads();

    float* dst = (b < PP) ? (out + 384 + 5*b) : (out + 1027);
    for (int it = 0; it < 5; ++it) {
        float m = 3.5e38f; int mi = -1;
        for (int i = threadIdx.x; i < 1280; i += 256)
            if (buf[i] < m) { m = buf[i]; mi = i; }
        rmin[threadIdx.x] = m; ridx[threadIdx.x] = mi;
        __syncthreads();
        for (int s = 128; s > 0; s >>= 1) {
            if (threadIdx.x < s && rmin[threadIdx.x + s] < rmin[threadIdx.x]) {
                rmin[threadIdx.x] = rmin[threadIdx.x + s];
                ridx[threadIdx.x] = ridx[threadIdx.x + s];
            }
            __syncthreads();
        }
        if (threadIdx.x == 0) {
            dst[it] = rmin[0];
            if (ridx[0] >= 0) buf[ridx[0]] = 3.6e38f;
        }
        __syncthreads();
    }
}

// ---------- Kernel 3: per-particle stats (closed form) + globals + DB index ----------
__global__ __launch_bounds__(128) void cnm_stats(
    const float* __restrict__ mx, const float* __restrict__ mq,
    const float* __restrict__ ws, float* __restrict__ out)
{
    __shared__ float srms[128];
    __shared__ float smx[384];
    __shared__ float smax[128];
    int p = threadIdx.x;
    float m0 = mx[3*p], m1 = mx[3*p+1], m2 = mx[3*p+2];
    smx[3*p] = m0; smx[3*p+1] = m1; smx[3*p+2] = m2;
    float pn2 = m0*m0 + m1*m1 + m2*m2;

    float Np = ws[p], S2 = ws[128 + p];
    float s10 = ws[256+3*p], s11 = ws[256+3*p+1], s12 = ws[256+3*p+2];
    float sq = ws[640], sq2 = ws[641], sq2xn = ws[642];
    float sx0 = ws[643], sx1 = ws[644], sx2 = ws[645], sxn2 = ws[646];

    float sum_dx2 = S2 - 2.0f*(m0*s10 + m1*s11 + m2*s12) + Np*pn2;
    float rms = sqrtf(sum_dx2 / Np);
    float mqv = mq[p];
    float sumq_mb = (float)PP * sq;  // mailbox quirk: sums entire [P,N] q mailbox
    float sum_dxq2 = mqv*mqv * (sq2xn - 2.0f*(m0*sx0 + m1*sx1 + m2*sx2) + pn2*sq2);
    float rmsq = sqrtf(sum_dxq2 / (Np * sumq_mb));

    out[p] = rms; out[128 + p] = rmsq; out[256 + p] = Np;
    srms[p] = rms;
    __syncthreads();

    float maxR = 0.0f;
    for (int j = 0; j < 128; ++j) {
        float d0 = m0 - smx[3*j], d1 = m1 - smx[3*j+1], d2 = m2 - smx[3*j+2];
        float M2 = d0*d0 + d1*d1 + d2*d2;
        float R = (M2 > 0.0f) ? (rms + srms[j]) / M2 : 0.0f;
        maxR = fmaxf(maxR, R);
    }
    smax[p] = maxR;
    __syncthreads();
    if (p == 0) {
        float s = 0.0f;
        for (int j = 0; j < 128; ++j) s += smax[j];
        out[1032] = s / 128.0f;                       // db
        out[1024] = sqrtf(sxn2 / (float)NN);          // rms_g
        out[1025] = sqrtf(sq2xn / ((float)NN * sq));  // rmsq_g
        out[1026] = (float)NN;                        // Ng
    }
}

// ---------- Kernel 4: N_node gather ----------
__global__ __launch_bounds__(256) void cnm_nnode(
    const int* __restrict__ parent, const float* __restrict__ ws, float* __restrict__ out)
{
    int n = blockIdx.x * 256 + threadIdx.x;
    out[1033 + n] = ws[parent[n]];
}

// ---------- Kernel 5: big [P,N] edge matrix via V_WMMA_F32_16X16X4_F32 ----------
// D = A x B with A[m] = (-2mx0,-2mx1,-2mx2,||mx||^2), B[.][c] = (x0,x1,x2,1)
// -> D[p,c] = ||mx_p||^2 - 2 mx_p . x_c ; add ||x_c||^2, sqrt -> distance.
__global__ __launch_bounds__(256) void cnm_edges(
    const float* __restrict__ x, const float* __restrict__ mx,
    const int* __restrict__ parent,
    float* __restrict__ dist, float* __restrict__ bel)
{
    const int lane  = threadIdx.x & 31;
    const int wave  = threadIdx.x >> 5;
    const int ct    = blockIdx.x * 8 + wave;   // column tile: 0..4095
    const int pt    = blockIdx.y;              // row tile: 0..7
    const int sub   = lane & 15;
    const int khalf = lane >> 4;
    const int col   = ct * 16 + sub;
    const int prow  = pt * 16 + sub;

    // A-matrix fragment (16x4 f32): lanes 0-15 hold K0/K1, lanes 16-31 hold K2/K3
    float m0 = mx[3*prow], m1 = mx[3*prow+1], m2 = mx[3*prow+2];
    float pn2 = m0*m0 + m1*m1 + m2*m2;
    v2f a;
    a.x = khalf ? (-2.0f * m2) : (-2.0f * m0);
    a.y = khalf ? pn2          : (-2.0f * m1);

    // B-matrix fragment (4x16 f32): same K split across half-waves
    float x0 = x[3*col], x1 = x[3*col+1], x2 = x[3*col+2];
    float xn2 = x0*x0 + x1*x1 + x2*x2;
    v2f b;
    b.x = khalf ? x2   : x0;
    b.y = khalf ? 1.0f : x1;

    v8f c = {};
    v8f d = __builtin_amdgcn_wmma_f32_16x16x4_f32(
        /*neg_a=*/false, a, /*neg_b=*/false, b,
        /*c_mod=*/(short)0, c, /*reuse_a=*/false, /*reuse_b=*/false);

    int par = parent[col];
    #pragma unroll
    for (int r = 0; r < 8; ++r) {
        int row = pt * 16 + r + 8 * khalf;    // D layout: VGPR r -> M=r (lanes<16) / M=r+8
        float dx2 = d[r] + xn2;
        float dv  = __builtin_amdgcn_sqrtf(fmaxf(dx2, 0.0f));  // raw v_sqrt_f32, no refinement
        int idx = row * NN + col;
        __builtin_nontemporal_store(dv, &dist[idx]);                       // streaming NT store
        __builtin_nontemporal_store((par == row) ? 1.0f : 0.0f, &bel[idx]);
    }
}

extern "C" void kernel_launch(void* const* d_in, const int* in_sizes, int n_in,
                              void* d_out, int out_size, void* d_ws, size_t ws_size,
                              hipStream_t stream) {
    const float* x       = (const float*)d_in[0];
    const float* q       = (const float*)d_in[1];
    const float* is_cond = (const float*)d_in[2];
    // d_in[3] beta, d_in[4] node_class: unused
    const float* mx      = (const float*)d_in[5];
    const float* mq      = (const float*)d_in[6];
    // d_in[7] particle_class: unused
    const int*   parent  = (const int*)d_in[8];
    // d_in[9] particle_idx: unused
    float* out = (float*)d_out;
    float* ws  = (float*)d_ws;

    hipMemsetAsync(ws, 0, 1024 * sizeof(float), stream);

    cnm_reduce<<<256, 256, 0, stream>>>(x, q, parent, ws);
    cnm_topk  <<<129, 256, 0, stream>>>(x, is_cond, mx, out);
    cnm_stats <<<1,   128, 0, stream>>>(mx, mq, ws, out);
    cnm_nnode <<<256, 256, 0, stream>>>(parent, ws, out);
    cnm_edges <<<dim3(512, 8), 256, 0, stream>>>(x, mx, parent,
                                                 out + 66569, out + 8455177);
}